// MPClayer_46729244180710
// MI455X (gfx1250) — compile-verified
//
#include <hip/hip_runtime.h>
#include <hip/hip_bf16.h>
#include <math.h>

// ---------------- problem constants ----------------
#define NH       256
#define NSTEP    64
#define BATCH    256
#define NHE      516                 // 2*(NH+2)
#define NHEP     528                 // NHE padded to multiple of 16
#define NU       128                 // 2*NSTEP
#define NCON     768                 // 12*NSTEP
#define NROWS    (NSTEP * NHE)       // 33024
#define DT_C     (1.0f / 60.0f)
#define EPS_C    1e-4f
#define PQ_C     5.0f
#define QV_C     200.0f
#define QA_C     1.0f
#define POS_MAX_C 70.0f
#define VEL_LIM_C 10.0f
#define ACC_LIM_C 100.0f
#define SIGMA_C  1e-6f
#define ADMM_ITERS_C 100

typedef float v2f __attribute__((ext_vector_type(2)));
typedef float v8f __attribute__((ext_vector_type(8)));

// ---------------- WMMA f32 16x16x4 ----------------
// A (16x4): lane<16 -> {K0,K1} of row lane ; lane>=16 -> {K2,K3} of row lane-16
// B (4x16): lane<16 -> {row0,row1} col lane ; lane>=16 -> {row2,row3} col lane-16
// C/D (16x16): vgpr v: lane<16 -> row v col lane ; lane>=16 -> row v+8 col lane-16
__device__ __forceinline__ v8f wmma_f32(v2f a, v2f b, v8f acc) {
  return __builtin_amdgcn_wmma_f32_16x16x4_f32(false, a, false, b, (short)0, acc,
                                               false, false);
}

__device__ __forceinline__ int clampi(int v, int hi) { return v < hi ? v : hi; }

// Guarded 64x16 store of 4 accumulators.
template <int LDC>
__device__ __forceinline__ void store4(float* __restrict__ C, int row0, int col0,
                                       int M, float alpha, int accum, v8f* acc) {
  const int lane = threadIdx.x & 31;
  const int c = lane & 15;
  const int rbase = (lane >> 4) << 3;
#pragma unroll
  for (int t = 0; t < 4; ++t) {
#pragma unroll
    for (int v = 0; v < 8; ++v) {
      int rr = row0 + 16 * t + rbase + v;
      if (rr < M) {
        size_t idx = (size_t)rr * LDC + col0 + c;
        float val = alpha * acc[t][v];
        if (accum) val += C[idx];
        C[idx] = val;
      }
    }
  }
}

// ---- 64x16 tile, C = alpha * A(MxK) @ B(KxN) (+C). Compile-time strides. ----
template <int LDA, int LDB, int LDC, int KK>
__device__ __forceinline__ void gemm4_nn(const float* __restrict__ A,
                                         const float* __restrict__ B,
                                         float* __restrict__ C,
                                         int row0, int col0, int M,
                                         float alpha, int accum) {
  const int lane = threadIdx.x & 31;
  const int r  = lane & 15;
  const int kk = (lane >> 4) << 1;  // 0 or 2
  const int c  = lane & 15;
  v8f acc[4] = {{}, {}, {}, {}};
  const float* Ar[4];
#pragma unroll
  for (int t = 0; t < 4; ++t)
    Ar[t] = A + (size_t)clampi(row0 + 16 * t + r, M - 1) * LDA;
  const float* Bp = B + col0 + c;
#pragma unroll 2
  for (int k0 = 0; k0 < KK; k0 += 4) {
    if (k0 + 32 < KK)
      __builtin_prefetch(&Bp[(size_t)(k0 + 32) * LDB], 0, 1);
    v2f b;
    b.x = Bp[(size_t)(k0 + kk) * LDB];
    b.y = Bp[(size_t)(k0 + kk + 1) * LDB];
#pragma unroll
    for (int t = 0; t < 4; ++t) {
      v2f a;
      a.x = Ar[t][k0 + kk];
      a.y = Ar[t][k0 + kk + 1];
      acc[t] = wmma_f32(a, b, acc[t]);
    }
  }
  store4<LDC>(C, row0, col0, M, alpha, accum, acc);
}

// ---- 64x16 tile, C = alpha * A(MxK) @ Bt(NxK)^T (+C) ----
template <int LDA, int LDBT, int LDC, int KK>
__device__ __forceinline__ void gemm4_nt(const float* __restrict__ A,
                                         const float* __restrict__ Bt,
                                         float* __restrict__ C,
                                         int row0, int col0, int M,
                                         float alpha, int accum) {
  const int lane = threadIdx.x & 31;
  const int r  = lane & 15;
  const int kk = (lane >> 4) << 1;
  const int c  = lane & 15;
  v8f acc[4] = {{}, {}, {}, {}};
  const float* Ar[4];
#pragma unroll
  for (int t = 0; t < 4; ++t)
    Ar[t] = A + (size_t)clampi(row0 + 16 * t + r, M - 1) * LDA;
  const float* Bcol = Bt + (size_t)(col0 + c) * LDBT;
#pragma unroll 2
  for (int k0 = 0; k0 < KK; k0 += 4) {
    v2f b;
    b.x = Bcol[k0 + kk];
    b.y = Bcol[k0 + kk + 1];
#pragma unroll
    for (int t = 0; t < 4; ++t) {
      v2f a;
      a.x = Ar[t][k0 + kk];
      a.y = Ar[t][k0 + kk + 1];
      acc[t] = wmma_f32(a, b, acc[t]);
    }
  }
  store4<LDC>(C, row0, col0, M, alpha, accum, acc);
}

// ---- 64x16 tile, C = alpha * At(KxM)^T @ B(KxN) (+C) ----
template <int LDAT, int LDB, int LDC, int KK>
__device__ __forceinline__ void gemm4_tn(const float* __restrict__ At,
                                         const float* __restrict__ B,
                                         float* __restrict__ C,
                                         int row0, int col0, int M,
                                         float alpha, int accum) {
  const int lane = threadIdx.x & 31;
  const int r  = lane & 15;
  const int kk = (lane >> 4) << 1;
  const int c  = lane & 15;
  v8f acc[4] = {{}, {}, {}, {}};
  const float* Ac[4];
#pragma unroll
  for (int t = 0; t < 4; ++t)
    Ac[t] = At + clampi(row0 + 16 * t + r, M - 1);
  const float* Bp = B + col0 + c;
#pragma unroll 2
  for (int k0 = 0; k0 < KK; k0 += 4) {
    v2f b;
    b.x = Bp[(size_t)(k0 + kk) * LDB];
    b.y = Bp[(size_t)(k0 + kk + 1) * LDB];
#pragma unroll
    for (int t = 0; t < 4; ++t) {
      v2f a;
      a.x = Ac[t][(size_t)(k0 + kk) * LDAT];
      a.y = Ac[t][(size_t)(k0 + kk + 1) * LDAT];
      acc[t] = wmma_f32(a, b, acc[t]);
    }
  }
  store4<LDC>(C, row0, col0, M, alpha, accum, acc);
}

// ------------- templated GEMM kernels (1 wave per block, 64x16/block) -------------
template <int LDA, int LDB, int LDC, int KK>
__global__ void k_gemm_nn(const float* A, const float* B, float* C, int M,
                          float alpha, int accum) {
  gemm4_nn<LDA, LDB, LDC, KK>(A, B, C, blockIdx.y << 6, blockIdx.x << 4, M, alpha,
                              accum);
}
template <int LDA, int LDBT, int LDC, int KK>
__global__ void k_gemm_nt(const float* A, const float* Bt, float* C, int M,
                          float alpha, int accum) {
  gemm4_nt<LDA, LDBT, LDC, KK>(A, Bt, C, blockIdx.y << 6, blockIdx.x << 4, M,
                               alpha, accum);
}
template <int LDAT, int LDB, int LDC, int KK>
__global__ void k_gemm_tn(const float* At, const float* B, float* C, int M,
                          float alpha, int accum) {
  gemm4_tn<LDAT, LDB, LDC, KK>(At, B, C, blockIdx.y << 6, blockIdx.x << 4, M,
                               alpha, accum);
}

// QS[k] = qscale[k] * Qblk @ S_block[k]   (batched over blockIdx.z)
__global__ void k_qs(const float* Qblk, const float* S, float* QS) {
  int k = blockIdx.z;
  float alpha = (k == NSTEP - 1) ? PQ_C : 1.0f;
  const float* B = S + (size_t)k * NHE * NU;
  float* C = QS + (size_t)k * NHE * NU;
  gemm4_nn<NHEP, NU, NU, NHE>(Qblk, B, C, blockIdx.y << 6, blockIdx.x << 4, NHE,
                              alpha, 0);
}

// ---------------- small build kernels ----------------
__global__ void k_zero(float* p, int n) {
  int i = blockIdx.x * blockDim.x + threadIdx.x;
  if (i < n) p[i] = 0.0f;
}
__global__ void k_copy(const float* s, float* d, int n) {
  int i = blockIdx.x * blockDim.x + threadIdx.x;
  if (i < n) d[i] = s[i];
}

__global__ void k_build_A0p(const float* Af, const float* Cf, float* A0p) {
  int idx = blockIdx.x * blockDim.x + threadIdx.x;
  if (idx >= NHEP * NHEP) return;
  int i = idx / NHEP, j = idx % NHEP;
  float v = 0.0f;
  if (i < NH) {
    if (j == i) v = 1.0f;
    else if (j == NH + 1) v = Af[i];
    else if (j == NHE - 1) v = Cf[i];
  } else if (i == NH) {
    if (j == NH) v = 1.0f;
    else if (j == NH + 1) v = DT_C;
  } else if (i == NH + 1) {
    if (j == NH + 1) v = 1.0f;
  } else if (i < NHE) {
    int r = i - (NH + 2);
    if (r < NH) {
      if (j == i) v = 1.0f;
      else if (j == NHE - 1) v = Af[r];
      else if (j == NH + 1) v = Cf[r];
    } else if (r == NH) {
      if (j == i) v = 1.0f;
      else if (j == i + 1) v = DT_C;
    } else {
      if (j == i) v = 1.0f;
    }
  }
  A0p[idx] = v;
}

__global__ void k_build_Qblk(const float* M1, const float* Mc, const float* alpha,
                             float* Qblk) {
  int idx = blockIdx.x * blockDim.x + threadIdx.x;
  if (idx >= NHEP * NHEP) return;
  int i = idx / NHEP, j = idx % NHEP;
  float v = 0.0f;
  if (i < NHE && j < NHE) {
    int bi = i / (NH + 2), bj = j / (NH + 2);
    int r = i - bi * (NH + 2), c = j - bj * (NH + 2);
    const float* Msrc = (bi == bj) ? M1 : Mc;
    float q = 0.0f;
    if (r < NH && c < NH) { q = Msrc[r * NH + c]; if (r == c) q += EPS_C; }
    else if (r == NH + 1 && c == NH + 1) q = QV_C;
    if (bi != bj) q *= alpha[0];
    v = q;
  }
  Qblk[idx] = v;
}

// S[i*NHE+r, 2j+s] = (A^{i-j} B0)[r,s] for j<=i
__global__ void k_build_S(const float* Tpack, float* S) {
  int idx = blockIdx.x * blockDim.x + threadIdx.x;
  if (idx >= NROWS * NU) return;
  int row = idx / NU, col = idx % NU;
  int i = row / NHE, r = row - i * NHE;
  int j = col >> 1, s = col & 1;
  int d = i - j;
  float v = 0.0f;
  const float hdt2 = 0.5f * DT_C * DT_C;
  if (d >= 0) {
    int c0 = s ? (NHE - 2) : NH;  // 514 or 256
    if (d == 0) {
      if (r == c0) v = hdt2;
      else if (r == c0 + 1) v = DT_C;
    } else {
      const float* Ad = Tpack + (size_t)(d - 1) * NHE * NHEP + (size_t)r * NHEP;
      v = hdt2 * Ad[c0] + DT_C * Ad[c0 + 1];
    }
  }
  S[idx] = v;
}

__global__ void k_fixup_qf(float* Qf) {
  int i = threadIdx.x;
  if (i < NU) Qf[i * NU + i] += 2.0f * QA_C + EPS_C;
}

__global__ void k_rho(const float* Qf, float* rho) {
  __shared__ float sm[NU];
  int t = threadIdx.x;
  sm[t] = Qf[t * NU + t];
  __syncthreads();
  for (int s = NU / 2; s > 0; s >>= 1) {
    if (t < s) sm[t] += sm[t + s];
    __syncthreads();
  }
  if (t == 0) *rho = sm[0] / (float)NU;
}

// G rows: [I, -I, S_pos, -S_pos, S_vel, -S_vel]; also writes GT.
__global__ void k_build_G(const float* S, float* G, float* GT) {
  int idx = blockIdx.x * blockDim.x + threadIdx.x;
  if (idx >= NCON * NU) return;
  int rr = idx / NU, cc = idx % NU;
  float v;
  if (rr < 2 * NU) {
    v = ((rr % NU) == cc) ? ((rr < NU) ? 1.0f : -1.0f) : 0.0f;
  } else {
    int seg = (rr - 2 * NU) / NU;  // 0:S_pos 1:-S_pos 2:S_vel 3:-S_vel
    int t = (rr - 2 * NU) % NU;
    int j = t & 63, agent = t >> 6;
    int srow = j * NHE + (agent ? (NHE - 2) : NH) + ((seg >= 2) ? 1 : 0);
    float sv = S[(size_t)srow * NU + cc];
    v = (seg & 1) ? -sv : sv;
  }
  G[(size_t)rr * NU + cc] = v;
  GT[(size_t)cc * NCON + rr] = v;
}

__global__ void k_build_Mtx(const float* Qf, const float* GtG, const float* rho,
                            float* Mtx) {
  int idx = blockIdx.x * blockDim.x + threadIdx.x;
  if (idx >= NU * NU) return;
  int i = idx / NU, j = idx % NU;
  Mtx[idx] = Qf[idx] + rho[0] * GtG[idx] + ((i == j) ? SIGMA_C : 0.0f);
}

// Gauss-Jordan with partial pivoting on [Mtx | I] (128x256), single workgroup.
__global__ void k_invert128(const float* Mtx, float* aug, float* Minv, float* MinvT) {
  const int n = NU, w = 2 * NU;
  int t = threadIdx.x;
  __shared__ int piv;
  __shared__ float fcol[NU];
  for (int idx = t; idx < n * w; idx += blockDim.x) {
    int i = idx / w, j = idx % w;
    aug[idx] = (j < n) ? Mtx[i * n + j] : (((j - n) == i) ? 1.0f : 0.0f);
  }
  __syncthreads();
  for (int p = 0; p < n; ++p) {
    if (t == 0) {
      int best = p;
      float bv = fabsf(aug[p * w + p]);
      for (int rr = p + 1; rr < n; ++rr) {
        float v = fabsf(aug[rr * w + p]);
        if (v > bv) { bv = v; best = rr; }
      }
      piv = best;
    }
    __syncthreads();
    int pr = piv;
    if (pr != p) {
      for (int j = t; j < w; j += blockDim.x) {
        float tmp = aug[p * w + j];
        aug[p * w + j] = aug[pr * w + j];
        aug[pr * w + j] = tmp;
      }
    }
    __syncthreads();
    for (int rr = t; rr < n; rr += blockDim.x) fcol[rr] = aug[rr * w + p];
    __syncthreads();
    float pinv = 1.0f / fcol[p];
    for (int j = t; j < w; j += blockDim.x) aug[p * w + j] *= pinv;
    __syncthreads();
    for (int idx = t; idx < n * w; idx += blockDim.x) {
      int rr = idx / w, j = idx % w;
      if (rr != p) aug[idx] -= fcol[rr] * aug[p * w + j];
    }
    __syncthreads();
  }
  for (int idx = t; idx < n * n; idx += blockDim.x) {
    int i = idx / n, j = idx % n;
    float v = aug[i * w + n + j];
    Minv[idx] = v;
    MinvT[j * n + i] = v;
  }
}

// x0p[b] = [x1(256), op, ov, x2(256), tp, tv, 0-pad(12)]
__global__ void k_build_x0p(const float* x1, const float* x2, const float* op,
                            const float* ov, const float* tp, const float* tv,
                            float* x0p) {
  int idx = blockIdx.x * blockDim.x + threadIdx.x;
  if (idx >= BATCH * NHEP) return;
  int b = idx / NHEP, col = idx % NHEP;
  float v = 0.0f;
  if (col < NH) v = x1[b * NH + col];
  else if (col == NH) v = op[b];
  else if (col == NH + 1) v = ov[b];
  else if (col < 2 * NH + 2) v = x2[b * NH + (col - NH - 2)];
  else if (col == 2 * NH + 2) v = tp[b];
  else if (col == 2 * NH + 3) v = tv[b];
  x0p[idx] = v;
}

// h[b] = [A, A, Pmax - Tpx, Tpx - Pmin, V - Tvx, V + Tvx]
__global__ void k_build_h(const float* x0p, const float* Tpack, float* h) {
  int b = blockIdx.x;
  int cth = threadIdx.x;  // 0..127
  int j = cth & 63, agent = cth >> 6;
  int prow = j * NHE + (agent ? (NHE - 2) : NH);
  const float* xr = x0p + (size_t)b * NHEP;
  const float* Tp = Tpack + (size_t)prow * NHEP;
  const float* Tv = Tp + NHEP;  // prow + 1
  float tpd = 0.0f, tvd = 0.0f;
  for (int k = 0; k < NHE; ++k) {
    float xv = xr[k];
    tpd += xv * Tp[k];
    tvd += xv * Tv[k];
  }
  float* hb = h + (size_t)b * NCON;
  hb[cth] = ACC_LIM_C;
  hb[NU + cth] = ACC_LIM_C;
  hb[2 * NU + cth] = POS_MAX_C - tpd;
  hb[3 * NU + cth] = tpd;  // -POS_MIN + Tpx, POS_MIN = 0
  hb[4 * NU + cth] = VEL_LIM_C - tvd;
  hb[5 * NU + cth] = VEL_LIM_C + tvd;
}

// ---------------- ADMM kernels (64x16 per wave) ----------------
// rhs = sigma*x - p + (rho*z - y) @ G    [A fused into loader]
__global__ void k_admm_rhs(const float* z, const float* y, const float* G,
                           const float* x, const float* p, const float* rho_ptr,
                           float* rhs) {
  const int lane = threadIdx.x & 31;
  const int col0 = blockIdx.x << 4, row0 = blockIdx.y << 6;
  const float rho = *rho_ptr;
  const int r = lane & 15, kk = (lane >> 4) << 1, c = lane & 15;
  v8f acc[4] = {{}, {}, {}, {}};
  const float *zr[4], *yr[4];
#pragma unroll
  for (int t = 0; t < 4; ++t) {
    zr[t] = z + (size_t)(row0 + 16 * t + r) * NCON;
    yr[t] = y + (size_t)(row0 + 16 * t + r) * NCON;
  }
  const float* Gp = G + col0 + c;
#pragma unroll 2
  for (int k0 = 0; k0 < NCON; k0 += 4) {
    v2f b;
    b.x = Gp[(size_t)(k0 + kk) * NU];
    b.y = Gp[(size_t)(k0 + kk + 1) * NU];
#pragma unroll
    for (int t = 0; t < 4; ++t) {
      v2f a;
      a.x = rho * zr[t][k0 + kk] - yr[t][k0 + kk];
      a.y = rho * zr[t][k0 + kk + 1] - yr[t][k0 + kk + 1];
      acc[t] = wmma_f32(a, b, acc[t]);
    }
  }
  const int rbase = (lane >> 4) << 3;
#pragma unroll
  for (int t = 0; t < 4; ++t) {
#pragma unroll
    for (int v = 0; v < 8; ++v) {
      int rr = row0 + 16 * t + rbase + v;
      size_t idx = (size_t)rr * NU + col0 + c;
      rhs[idx] = acc[t][v] + SIGMA_C * x[idx] - p[idx];
    }
  }
}

// Gx = x @ G^T (via GT), then z = min(Gx + y/rho, h); y += rho*(Gx - z)
__global__ void k_admm_zy(const float* x, const float* GT, const float* h,
                          const float* rho_ptr, float* z, float* y) {
  const int lane = threadIdx.x & 31;
  const int col0 = blockIdx.x << 4, row0 = blockIdx.y << 6;
  const float rho = *rho_ptr, rinv = 1.0f / *rho_ptr;
  const int r = lane & 15, kk = (lane >> 4) << 1, c = lane & 15;
  v8f acc[4] = {{}, {}, {}, {}};
  const float* Ar[4];
#pragma unroll
  for (int t = 0; t < 4; ++t) Ar[t] = x + (size_t)(row0 + 16 * t + r) * NU;
  const float* Gp = GT + col0 + c;
#pragma unroll 2
  for (int k0 = 0; k0 < NU; k0 += 4) {
    v2f b;
    b.x = Gp[(size_t)(k0 + kk) * NCON];
    b.y = Gp[(size_t)(k0 + kk + 1) * NCON];
#pragma unroll
    for (int t = 0; t < 4; ++t) {
      v2f a;
      a.x = Ar[t][k0 + kk];
      a.y = Ar[t][k0 + kk + 1];
      acc[t] = wmma_f32(a, b, acc[t]);
    }
  }
  const int rbase = (lane >> 4) << 3;
#pragma unroll
  for (int t = 0; t < 4; ++t) {
#pragma unroll
    for (int v = 0; v < 8; ++v) {
      int rr = row0 + 16 * t + rbase + v;
      size_t idx = (size_t)rr * NCON + col0 + c;
      float gx = acc[t][v];
      float zn = fminf(gx + y[idx] * rinv, h[idx]);
      y[idx] = y[idx] + rho * (gx - zn);
      z[idx] = zn;
    }
  }
}

// ---------------- host driver ----------------
extern "C" void kernel_launch(void* const* d_in, const int* in_sizes, int n_in,
                              void* d_out, int out_size, void* d_ws, size_t ws_size,
                              hipStream_t stream) {
  (void)in_sizes; (void)n_in; (void)out_size; (void)ws_size;
  const float* x1  = (const float*)d_in[0];
  const float* x2  = (const float*)d_in[1];
  const float* op  = (const float*)d_in[2];
  const float* ov  = (const float*)d_in[3];
  const float* tp  = (const float*)d_in[4];
  const float* tv  = (const float*)d_in[5];
  const float* Af  = (const float*)d_in[6];
  const float* Cf  = (const float*)d_in[7];
  const float* Lq  = (const float*)d_in[8];
  const float* Lqc = (const float*)d_in[9];
  const float* alp = (const float*)d_in[10];
  float* out = (float*)d_out;

  // ---- carve workspace (floats, 256B-aligned chunks); total ~112 MB ----
  float* w = (float*)d_ws;
  auto alloc = [&](size_t n) { float* p = w; w += (n + 63) & ~(size_t)63; return p; };
  float* A0p   = alloc((size_t)NHEP * NHEP);
  float* M1    = alloc((size_t)NH * NH);
  float* Mc    = alloc((size_t)NH * NH);
  float* Qblk  = alloc((size_t)NHEP * NHEP);
  float* Tpack = alloc((size_t)NSTEP * NHE * NHEP);  // A^1..A^64, 516x528 each
  float* S     = alloc((size_t)NROWS * NU);
  float* QS    = alloc((size_t)NROWS * NU);
  float* Qf    = alloc((size_t)NU * NU);
  float* pf    = alloc((size_t)NHEP * NU);
  float* G     = alloc((size_t)NCON * NU);
  float* GT    = alloc((size_t)NU * NCON);
  float* GtG   = alloc((size_t)NU * NU);
  float* Mtx   = alloc((size_t)NU * NU);
  float* aug   = alloc((size_t)NU * 2 * NU);
  float* Minv  = alloc((size_t)NU * NU);
  float* MinvT = alloc((size_t)NU * NU);
  float* x0p   = alloc((size_t)BATCH * NHEP);
  float* pvec  = alloc((size_t)BATCH * NU);
  float* h     = alloc((size_t)BATCH * NCON);
  float* xv    = alloc((size_t)BATCH * NU);
  float* zv    = alloc((size_t)BATCH * NCON);
  float* yv    = alloc((size_t)BATCH * NCON);
  float* rhs   = alloc((size_t)BATCH * NU);
  float* rho   = alloc((size_t)64);

  const dim3 b32(32), b128(128), b256(256);
  auto cdiv = [](int a, int b) { return (a + b - 1) / b; };

  // ---- static model build ----
  k_build_A0p<<<cdiv(NHEP * NHEP, 256), b256, 0, stream>>>(Af, Cf, A0p);
  // M1 = Lq Lq^T ; Mc = Lqc Lqc^T
  k_gemm_nt<NH, NH, NH, NH><<<dim3(NH / 16, NH / 64), b32, 0, stream>>>(
      Lq, Lq, M1, NH, 1.0f, 0);
  k_gemm_nt<NH, NH, NH, NH><<<dim3(NH / 16, NH / 64), b32, 0, stream>>>(
      Lqc, Lqc, Mc, NH, 1.0f, 0);
  k_build_Qblk<<<cdiv(NHEP * NHEP, 256), b256, 0, stream>>>(M1, Mc, alp, Qblk);

  // ---- matrix power chain: Tpack[k] = A^{k+1} (516x528 blocks) ----
  k_copy<<<cdiv(NHE * NHEP, 256), b256, 0, stream>>>(A0p, Tpack, NHE * NHEP);
  for (int k = 1; k < NSTEP; ++k) {
    k_gemm_nn<NHEP, NHEP, NHEP, NHE>
        <<<dim3(NHEP / 16, cdiv(NHE, 64)), b32, 0, stream>>>(
            A0p, Tpack + (size_t)(k - 1) * NHE * NHEP,
            Tpack + (size_t)k * NHE * NHEP, NHE, 1.0f, 0);
  }

  // ---- S, QS, Qf, pf ----
  k_build_S<<<cdiv(NROWS * NU, 256), b256, 0, stream>>>(Tpack, S);
  k_qs<<<dim3(NU / 16, cdiv(NHE, 64), NSTEP), b32, 0, stream>>>(Qblk, S, QS);
  k_gemm_tn<NU, NU, NU, NROWS><<<dim3(NU / 16, NU / 64), b32, 0, stream>>>(
      S, QS, Qf, NU, 2.0f, 0);
  k_fixup_qf<<<1, b128, 0, stream>>>(Qf);
  k_gemm_tn<NHEP, NU, NU, NROWS><<<dim3(NU / 16, cdiv(NHEP, 64)), b32, 0, stream>>>(
      Tpack, QS, pf, NHEP, 2.0f, 0);

  // ---- constraints, rho, M^-1 ----
  k_build_G<<<cdiv(NCON * NU, 256), b256, 0, stream>>>(S, G, GT);
  k_gemm_tn<NU, NU, NU, NCON><<<dim3(NU / 16, NU / 64), b32, 0, stream>>>(
      G, G, GtG, NU, 1.0f, 0);
  k_rho<<<1, b128, 0, stream>>>(Qf, rho);
  k_build_Mtx<<<cdiv(NU * NU, 256), b256, 0, stream>>>(Qf, GtG, rho, Mtx);
  k_invert128<<<1, b256, 0, stream>>>(Mtx, aug, Minv, MinvT);

  // ---- batch-dependent setup ----
  k_build_x0p<<<cdiv(BATCH * NHEP, 256), b256, 0, stream>>>(x1, x2, op, ov, tp, tv,
                                                            x0p);
  k_gemm_nn<NHEP, NU, NU, NHEP><<<dim3(NU / 16, BATCH / 64), b32, 0, stream>>>(
      x0p, pf, pvec, BATCH, 1.0f, 0);
  k_build_h<<<BATCH, b128, 0, stream>>>(x0p, Tpack, h);
  k_zero<<<cdiv(BATCH * NU, 256), b256, 0, stream>>>(xv, BATCH * NU);
  k_zero<<<cdiv(BATCH * NCON, 256), b256, 0, stream>>>(zv, BATCH * NCON);
  k_zero<<<cdiv(BATCH * NCON, 256), b256, 0, stream>>>(yv, BATCH * NCON);

  // ---- ADMM: 100 iterations, 3 WMMA kernels each ----
  for (int it = 0; it < ADMM_ITERS_C; ++it) {
    k_admm_rhs<<<dim3(NU / 16, BATCH / 64), b32, 0, stream>>>(zv, yv, G, xv, pvec,
                                                              rho, rhs);
    k_gemm_nn<NU, NU, NU, NU><<<dim3(NU / 16, BATCH / 64), b32, 0, stream>>>(
        rhs, MinvT, xv, BATCH, 1.0f, 0);
    k_admm_zy<<<dim3(NCON / 16, BATCH / 64), b32, 0, stream>>>(xv, GT, h, rho, zv,
                                                               yv);
  }

  // ---- x_pred = u @ S^T + x0 @ T^T ----
  k_gemm_nt<NU, NU, NROWS, NU><<<dim3(NROWS / 16, BATCH / 64), b32, 0, stream>>>(
      xv, S, out, BATCH, 1.0f, 0);
  k_gemm_nt<NHEP, NHEP, NROWS, NHEP>
      <<<dim3(NROWS / 16, BATCH / 64), b32, 0, stream>>>(x0p, Tpack, out, BATCH,
                                                         1.0f, 1);
}